// SpatialVLMEncoder_55576876810961
// MI455X (gfx1250) — compile-verified
//
#include <hip/hip_runtime.h>
#include <hip/hip_bf16.h>
#include <stdint.h>

// Problem constants from the reference.
#define Bc 8
#define Lc 2048
#define Pc 576
#define Vc 32000
#define Dc 2560
#define Tc (Lc - 1 + Pc)      // 2623
#define ROW_BYTES (Dc * 4)    // 10240
#define IGNORE_INDEX_F (-100.0f)

// Row copy: one workgroup per TWO output rows (b, t0) and (b, t0+1).
// 320 threads = 10 wave32s; each lane moves two 16-byte chunks per row
// (320 * 16 * 2 = 10240 B = one row exactly).
// Copy path: global -> LDS -> global with CDNA5 async LDS transfer ops
// (ASYNCcnt), uniform SGPR base (GVS mode) + per-lane 32-bit offset.
// All loads are issued before the single s_wait_asynccnt 0 (4 b128 loads in
// flight per lane), then all stores issue with a non-temporal hint so the
// write-once 215 MB output does not evict reusable embed-table rows from L2.
__global__ __launch_bounds__(320) void splice_rows_async(
    const float* __restrict__ embed_table,     // [V, D]
    const float* __restrict__ image_features,  // [B, P, D]
    const int*   __restrict__ input_ids,       // [B, L]
    const int*   __restrict__ img_pos,         // [B]
    float*       __restrict__ out)             // [B, T, D]
{
    __shared__ __align__(16) float smem[2 * Dc];   // 20480 B staging (2 rows)

    const int t0 = blockIdx.x * 2;  // first row handled by this block
    const int b  = blockIdx.y;
    const bool has2 = (t0 + 1) < Tc;            // uniform across the block

    const int pos = img_pos[b];                 // uniform -> scalar load

    // Resolve the uniform source row pointer for output row t.
    auto src_for = [&](int t) -> const float* {
        const bool is_img = (t >= pos) && (t < pos + Pc);
        if (is_img) {
            return image_features + ((size_t)b * Pc + (size_t)(t - pos)) * Dc;
        }
        int tok = (t < pos) ? t : (t - Pc + 1);
        tok = tok < 0 ? 0 : (tok > (Lc - 1) ? (Lc - 1) : tok);
        const int id = input_ids[b * Lc + tok]; // uniform -> scalar load
        return embed_table + (size_t)id * Dc;
    };

    const float* src0 = src_for(t0);
    const float* src1 = has2 ? src_for(t0 + 1) : src0;
    float* dst0 = out + ((size_t)b * Tc + (size_t)t0) * Dc;
    float* dst1 = dst0 + Dc;

    // Per-lane byte offset within a row; second chunk is +5120 via inst offset.
    const uint32_t off0 = (uint32_t)threadIdx.x * 16u;
    // Workgroup-relative LDS byte address (low 32 bits of the shared-aperture
    // flat pointer are the LDS offset the async VDST field expects).
    const uint32_t lds0 = (uint32_t)(uintptr_t)(&smem[0]) + off0;
    const uint32_t lds1 = lds0 + ROW_BYTES;

    // ---- Phase 1: issue ALL async loads (global -> LDS), 4 b128 per lane.
    // The instruction immediate offset is added to BOTH the LDS address and
    // the memory address, so one offset covers both sides of the transfer.
    asm volatile(
        "global_load_async_to_lds_b128 %0, %1, %2\n\t"
        "global_load_async_to_lds_b128 %0, %1, %2 offset:5120"
        :
        : "v"(lds0), "v"(off0), "s"(src0)
        : "memory");
    if (has2) {
        asm volatile(
            "global_load_async_to_lds_b128 %0, %1, %2\n\t"
            "global_load_async_to_lds_b128 %0, %1, %2 offset:5120"
            :
            : "v"(lds1), "v"(off0), "s"(src1)
            : "memory");
    }

    // Loads and stores on the shared ASYNCcnt complete out of order w.r.t.
    // each other, so all loads must drain before a store reads that LDS.
    asm volatile("s_wait_asynccnt 0" ::: "memory");

    // ---- Phase 2: issue ALL async stores (LDS -> global), non-temporal.
    // Same lane<->LDS mapping as the loads: no cross-wave barrier needed.
    asm volatile(
        "global_store_async_from_lds_b128 %1, %0, %2 th:TH_STORE_NT\n\t"
        "global_store_async_from_lds_b128 %1, %0, %2 offset:5120 th:TH_STORE_NT"
        :
        : "v"(lds0), "v"(off0), "s"(dst0)
        : "memory");
    if (has2) {
        asm volatile(
            "global_store_async_from_lds_b128 %1, %0, %2 th:TH_STORE_NT\n\t"
            "global_store_async_from_lds_b128 %1, %0, %2 offset:5120 th:TH_STORE_NT"
            :
            : "v"(lds1), "v"(off0), "s"(dst1)
            : "memory");
    }
    asm volatile("s_wait_asynccnt 0" ::: "memory");
}

// Metadata: new_labels (as float), attention_mask (1.0f), position_ids (t).
// d_out tail layout: [labels B*T][mask B*T][position_ids B*T], all float32.
__global__ void splice_meta(
    const int* __restrict__ labels,    // [B, L]
    const int* __restrict__ img_pos,   // [B]
    float*     __restrict__ out_meta)  // [3 * B * T]
{
    const int idx = blockIdx.x * blockDim.x + threadIdx.x;
    if (idx >= Bc * Tc) return;
    const int b = idx / Tc;
    const int t = idx - b * Tc;

    const int pos = img_pos[b];
    const bool is_img = (t >= pos) && (t < pos + Pc);

    int tok = (t < pos) ? t : (t - Pc + 1);
    tok = tok < 0 ? 0 : (tok > (Lc - 1) ? (Lc - 1) : tok);

    float lab = is_img ? IGNORE_INDEX_F : (float)labels[b * Lc + tok];

    out_meta[idx] = lab;                              // new_labels
    out_meta[(size_t)Bc * Tc + idx] = 1.0f;           // attention_mask
    out_meta[(size_t)2 * Bc * Tc + idx] = (float)t;   // position_ids
}

extern "C" void kernel_launch(void* const* d_in, const int* in_sizes, int n_in,
                              void* d_out, int out_size, void* d_ws, size_t ws_size,
                              hipStream_t stream) {
    // setup_inputs() order: embed_table, image_features, input_ids, labels, img_pos
    const float* embed_table    = (const float*)d_in[0];
    const float* image_features = (const float*)d_in[1];
    const int*   input_ids      = (const int*)d_in[2];
    const int*   labels         = (const int*)d_in[3];
    const int*   img_pos        = (const int*)d_in[4];

    float* out = (float*)d_out;
    float* out_meta = out + (size_t)Bc * Tc * Dc;

    dim3 grid((Tc + 1) / 2, Bc);   // 1312 x 8 blocks, 2 rows each
    splice_rows_async<<<grid, 320, 0, stream>>>(
        embed_table, image_features, input_ids, img_pos, out);

    const int n = Bc * Tc;
    splice_meta<<<(n + 255) / 256, 256, 0, stream>>>(labels, img_pos, out_meta);
}